// PolyHarmonicOscillator_13804024890037
// MI455X (gfx1250) — compile-verified
//
#include <hip/hip_runtime.h>
#include <math.h>

// PolyHarmonicOscillator for MI455X (gfx1250, wave32, WMMA).
//
// Kernel 1: per (batch, oscillator) double-precision frame-phase prefix
//           (closed-form frame advance 8*gp + 48*gc + 8*gn), stored mod 1.
// Kernel 2: per (batch, frame) block; 8 waves; oscillator reduction done with
//           V_WMMA_F32_16X16X4_F32 against an all-ones B matrix (f32 exact).
//           sin(2*pi*ph) computed with hardware v_fract_f32 + v_sin_f32
//           (V_SIN_F32 input is in rotations: D = sin(2*pi*S0)).

typedef __attribute__((ext_vector_type(2))) float v2f;
typedef __attribute__((ext_vector_type(8))) float v8f;

#define SRf     16000.0f
#define NYQ     8000.0f
#define FRAMES  500
#define AUDIO   (FRAMES * 64)
#define NOSC    512          // poly(8) * harm(64)
#define BATCH   2

__device__ __forceinline__ float sin2pi(float x) {
    // sin is 1-periodic in rotations -> fract() is lossless and keeps the
    // argument inside v_sin_f32's valid domain.
    return __builtin_amdgcn_sinf(__builtin_amdgcn_fractf(x));
}

__global__ __launch_bounds__(512)
void phk_base(const float* __restrict__ f0,        // (2,8,500)
              const float* __restrict__ rp,        // (2,512,1)
              float* __restrict__ base)            // ws: (2,512,500)
{
    int idx = blockIdx.x * blockDim.x + threadIdx.x;   // 0..1023
    if (idx >= BATCH * NOSC) return;
    int b = idx >> 9;
    int o = idx & (NOSC - 1);
    int p = o >> 6;
    int h = o & 63;
    double scale = (double)(h + 1) / 16000.0;          // Hz -> cycles/sample
    const float* f  = f0 + (b * 8 + p) * FRAMES;
    float* out      = base + (size_t)idx * FRAMES;
    double acc = (double)rp[idx];                      // random phase offset
    for (int t = 0; t < FRAMES; ++t) {
        out[t] = (float)(acc - floor(acc));            // frame-start phase mod 1
        double gp = (double)f[t > 0 ? t - 1 : 0] * scale;
        double gc = (double)f[t] * scale;
        double gn = (double)f[t < FRAMES - 1 ? t + 1 : FRAMES - 1] * scale;
        acc += 8.0 * gp + 48.0 * gc + 8.0 * gn;        // exact full-frame advance
    }
}

__global__ __launch_bounds__(256)
void phk_synth(const float* __restrict__ f0,   // (2,8,500)
               const float* __restrict__ c,    // (2,8,64,500)
               const float* __restrict__ v,    // (2,8,500)
               const float* __restrict__ base, // ws (2,512,500)
               float* __restrict__ out)        // (2,32000)
{
    __shared__ float base_s[NOSC];
    __shared__ float g_s[NOSC * 3];   // gprev, gcur, gnext (cycles/sample)
    __shared__ float c_s[NOSC * 3];   // nyquist-masked amplitude frames
    __shared__ float v_s[NOSC * 3];
    __shared__ float pout[2][64];     // per-partition partial sums

    int bid = blockIdx.x;
    int t = bid % FRAMES;
    int b = bid / FRAMES;
    int tid = threadIdx.x;
    int tm = (t > 0) ? t - 1 : 0;
    int tp = (t < FRAMES - 1) ? t + 1 : FRAMES - 1;

    // ---- stage per-oscillator frame data into LDS ----
    for (int o = tid; o < NOSC; o += 256) {
        int p = o >> 6, h = o & 63;
        float hm = (float)(h + 1);
        const float* fptr = f0 + (b * 8 + p) * FRAMES;
        float fbm = fptr[tm] * hm;                 // f0_bank at t-1 (Hz)
        float fbc = fptr[t]  * hm;
        float fbp = fptr[tp] * hm;
        g_s[o * 3 + 0] = fbm / SRf;
        g_s[o * 3 + 1] = fbc / SRf;
        g_s[o * 3 + 2] = fbp / SRf;
        const float* cptr = c + ((b * 8 + p) * 64 + h) * FRAMES;
        c_s[o * 3 + 0] = (fbm < NYQ) ? cptr[tm] : 0.0f;   // mask before upsample
        c_s[o * 3 + 1] = (fbc < NYQ) ? cptr[t]  : 0.0f;
        c_s[o * 3 + 2] = (fbp < NYQ) ? cptr[tp] : 0.0f;
        const float* vptr = v + (b * 8 + p) * FRAMES;
        v_s[o * 3 + 0] = vptr[tm];
        v_s[o * 3 + 1] = vptr[t];
        v_s[o * 3 + 2] = vptr[tp];
        base_s[o] = base[(size_t)(b * NOSC + o) * FRAMES + t];
    }
    __syncthreads();

    // ---- wave mapping: 8 waves = 4 sample-groups x 2 oscillator partitions ----
    int wave = tid >> 5;
    int lane = tid & 31;
    int g16  = wave & 3;          // sample group: samples [g16*16, g16*16+16)
    int part = wave >> 2;         // oscillator partition: [part*256, part*256+256)
    int m = g16 * 16 + (lane & 15);   // sample index within frame, 0..63

    // Closed-form inclusive within-frame phase:  K0*gprev + K1*gcur + K2*gnext
    float K0, K1, K2, w;
    int half;
    if (m < 32) {                 // interpolates between frames t-1 and t
        float mm = (float)m;
        float A2 = (mm + 1.0f) * (mm + 65.0f) * (1.0f / 128.0f);
        K0 = (mm + 1.0f) - A2;  K1 = A2;  K2 = 0.0f;
        w = (mm + 32.5f) * (1.0f / 64.0f);
        half = 0;
    } else {                      // interpolates between frames t and t+1
        float mm2 = (float)(m - 31);
        float B2 = mm2 * mm2 * (1.0f / 128.0f);
        K0 = 8.0f;  K1 = 24.0f + mm2 - B2;  K2 = B2;
        w = (mm2 - 0.5f) * (1.0f / 64.0f);
        half = 1;
    }
    int koff = (lane >> 4) << 1;  // lanes 0-15 -> K=0,1 ; lanes 16-31 -> K=2,3

    v2f ones; ones.x = 1.0f; ones.y = 1.0f;     // B matrix: all ones (layout-proof)
    v8f acc = {};

    int obase = part * 256;
#pragma unroll 4
    for (int kk = 0; kk < 64; ++kk) {
        int o0 = obase + kk * 4 + koff;
        float s0, s1;
        {
            int o = o0;
            float ph  = base_s[o] + K0 * g_s[o * 3] + K1 * g_s[o * 3 + 1]
                                  + K2 * g_s[o * 3 + 2];
            float ca = c_s[o * 3 + half], cb = c_s[o * 3 + half + 1];
            float va = v_s[o * 3 + half], vb = v_s[o * 3 + half + 1];
            float amp = (ca + w * (cb - ca)) * (va + w * (vb - va));
            s0 = 0.04f * amp * sin2pi(ph);
        }
        {
            int o = o0 + 1;
            float ph  = base_s[o] + K0 * g_s[o * 3] + K1 * g_s[o * 3 + 1]
                                  + K2 * g_s[o * 3 + 2];
            float ca = c_s[o * 3 + half], cb = c_s[o * 3 + half + 1];
            float va = v_s[o * 3 + half], vb = v_s[o * 3 + half + 1];
            float amp = (ca + w * (cb - ca)) * (va + w * (vb - va));
            s1 = 0.04f * amp * sin2pi(ph);
        }
        v2f A; A.x = s0; A.y = s1;
        // D[m][n] += sum_k A[m][k]  (B all ones) — full f32 precision reduction
        acc = __builtin_amdgcn_wmma_f32_16x16x4_f32(
            false, A, false, ones, (short)0, acc, false, false);
    }

    // Extract column 0 of D: lanes 0/16 hold M = r / r+8 in acc[r].
    if (lane == 0) {
#pragma unroll
        for (int r = 0; r < 8; ++r) pout[part][g16 * 16 + r] = acc[r];
    }
    if (lane == 16) {
#pragma unroll
        for (int r = 0; r < 8; ++r) pout[part][g16 * 16 + 8 + r] = acc[r];
    }
    __syncthreads();

    if (tid < 64) {
        out[(size_t)b * AUDIO + t * 64 + tid] = pout[0][tid] + pout[1][tid];
    }
}

extern "C" void kernel_launch(void* const* d_in, const int* in_sizes, int n_in,
                              void* d_out, int out_size, void* d_ws, size_t ws_size,
                              hipStream_t stream) {
    (void)in_sizes; (void)n_in; (void)out_size; (void)ws_size;
    const float* f0 = (const float*)d_in[0];   // (2,8,500)
    const float* c  = (const float*)d_in[1];   // (2,8,64,500)
    const float* v  = (const float*)d_in[2];   // (2,8,500)
    /* d_in[3] = a — unused by the reference output */
    const float* rp = (const float*)d_in[4];   // (2,512,1)

    float* base = (float*)d_ws;                // 2*512*500 floats = 2 MB

    phk_base<<<2, 512, 0, stream>>>(f0, rp, base);
    phk_synth<<<BATCH * FRAMES, 256, 0, stream>>>(f0, c, v, base, (float*)d_out);
}